// sp_attn_head_26963804684998
// MI455X (gfx1250) — compile-verified
//
#include <hip/hip_runtime.h>
#include <hip/hip_bf16.h>

typedef __attribute__((ext_vector_type(2))) float v2f;
typedef __attribute__((ext_vector_type(8))) float v8f;

#define F_IN   256
#define F_OUT  64
#define NEG_SLOPE 0.2f

// ---------------------------------------------------------------------------
// order-preserving float <-> uint mapping for atomic float-max
// ---------------------------------------------------------------------------
__device__ __forceinline__ unsigned f2sortable(float f) {
    unsigned u = __float_as_uint(f);
    return (u & 0x80000000u) ? ~u : (u | 0x80000000u);
}
__device__ __forceinline__ float sortable2f(unsigned u) {
    return __uint_as_float((u & 0x80000000u) ? (u ^ 0x80000000u) : ~u);
}

// ---------------------------------------------------------------------------
// 0) deterministic init: zero accumulator (d_out), row-max, denom
// ---------------------------------------------------------------------------
__global__ void init_kernel(float* __restrict__ acc, unsigned* __restrict__ rmax,
                            float* __restrict__ denom, int n) {
    int i = blockIdx.x * blockDim.x + threadIdx.x;
    if (i < n * F_OUT) acc[i] = 0.0f;
    if (i < n) { rmax[i] = 0u; denom[i] = 0.0f; }
}

// ---------------------------------------------------------------------------
// 1) seq_fts = seq @ W via V_WMMA_F32_16X16X4_F32, fused f1/f2 epilogue.
//    block = 256 threads = 8 waves; each wave owns a 16x64 row-stripe:
//    4 accumulators (one per 16-col tile), so each A fragment is loaded from
//    global exactly once and reused 4x from registers (minimal HBM traffic,
//    4 independent WMMAs in flight per k-step).
//    W is staged in LDS pre-swizzled as K-pairs:
//        Wl[(kk/2)*128 + col*2 + {0,1}] = { W[kk][col], W[kk+1][col] }
//    so each lane's B fragment is a single contiguous ds_load_b64 into an
//    even-aligned VGPR pair (no repacking moves before the WMMA).
// ---------------------------------------------------------------------------
__global__ void __launch_bounds__(256)
gemm_wmma_kernel(const float* __restrict__ seq, const float* __restrict__ W,
                 float* __restrict__ fts,
                 const float* __restrict__ a1, const float* __restrict__ b1,
                 const float* __restrict__ a2, const float* __restrict__ b2,
                 float* __restrict__ f1, float* __restrict__ f2, int n) {
    __shared__ float Wl[F_IN * F_OUT];               // 64 KB, whole W in LDS
    // cooperative load + K-pair swizzle: W[kk][col] -> Wl[(kk>>1)*128 + col*2 + (kk&1)]
    for (int i = threadIdx.x; i < F_IN * F_OUT; i += 256) {
        const int kk  = i >> 6;          // 0..255
        const int col = i & 63;
        Wl[(kk >> 1) * (2 * F_OUT) + col * 2 + (kk & 1)] = W[i];
    }
    __syncthreads();

    const int wave = threadIdx.x >> 5;
    const int lane = threadIdx.x & 31;
    const int rowTile = blockIdx.x * 8 + wave;       // 16 rows per wave

    int row = rowTile * 16 + (lane & 15);
    if (row >= n) row = n - 1;                       // clamp: keep EXEC all-1s
    const int lcol  = lane & 15;                     // col within a 16-col tile
    const int khalf = (lane >> 4) * 2;               // 0 for lanes 0-15, 2 for 16-31

    const float* __restrict__ arow = seq + (size_t)row * F_IN;

    v8f acc[4] = {v8f{}, v8f{}, v8f{}, v8f{}};
    for (int k = 0; k < F_IN; k += 4) {
        // A fragment (16x4 f32): vgpr0 = K = k+khalf, vgpr1 = K = k+khalf+1
        float2 av = *(const float2*)(arow + k + khalf);
        v2f a; a[0] = av.x; a[1] = av.y;
        const int kpairBase = ((k + khalf) >> 1) * (2 * F_OUT);
#pragma unroll
        for (int ct = 0; ct < 4; ++ct) {
            // B fragment (4x16 f32): contiguous K-pair in swizzled LDS
            float2 bv = *(const float2*)(&Wl[kpairBase + (ct * 16 + lcol) * 2]);
            v2f b; b[0] = bv.x; b[1] = bv.y;
            acc[ct] = __builtin_amdgcn_wmma_f32_16x16x4_f32(
                    /*neg_a=*/false, a, /*neg_b=*/false, b,
                    /*c_mod=*/(short)0, acc[ct], /*reuse_a=*/false, /*reuse_b=*/false);
        }
    }

    // --- store seq_fts tiles: C/D layout VGPR r -> M = r + 8*(lane>=16) ---
    const int nhi = lane >> 4;
#pragma unroll
    for (int ct = 0; ct < 4; ++ct) {
        const int ocol = ct * 16 + lcol;
#pragma unroll
        for (int r = 0; r < 8; ++r) {
            const int orow = rowTile * 16 + r + 8 * nhi;
            if (orow < n) fts[(size_t)orow * F_OUT + ocol] = acc[ct][r];
        }
    }

    // --- fused f1/f2 epilogue from live accumulators ---
    float A1[4], A2[4];
#pragma unroll
    for (int ct = 0; ct < 4; ++ct) {
        A1[ct] = a1[ct * 16 + lcol];
        A2[ct] = a2[ct * 16 + lcol];
    }
    const float b1v = b1[0], b2v = b2[0];
#pragma unroll
    for (int r = 0; r < 8; ++r) {
        float p1 = 0.0f, p2 = 0.0f;
#pragma unroll
        for (int ct = 0; ct < 4; ++ct) {
            p1 += acc[ct][r] * A1[ct];
            p2 += acc[ct][r] * A2[ct];
        }
        // reduce across the 16 lanes of this row (xor 8..1 stays in-half)
#pragma unroll
        for (int off = 8; off > 0; off >>= 1) {
            p1 += __shfl_xor(p1, off, 32);
            p2 += __shfl_xor(p2, off, 32);
        }
        if ((lane & 15) == 0) {
            const int orow = rowTile * 16 + r + 8 * nhi;
            if (orow < n) { f1[orow] = p1 + b1v; f2[orow] = p2 + b2v; }
        }
    }
}

// ---------------------------------------------------------------------------
// 2) edge logits + leaky-relu + segment max (atomic on sortable-uint)
// ---------------------------------------------------------------------------
__global__ void logits_kernel(const int* __restrict__ erow, const int* __restrict__ ecol,
                              const float* __restrict__ f1, const float* __restrict__ f2,
                              float* __restrict__ logits, unsigned* __restrict__ rmax, int e) {
    const int i = blockIdx.x * blockDim.x + threadIdx.x;
    if (i >= e) return;
    const int r = erow[i], c = ecol[i];
    float l = f1[r] + f2[c];
    l = l > 0.0f ? l : NEG_SLOPE * l;
    logits[i] = l;
    atomicMax(&rmax[r], f2sortable(l));
}

// ---------------------------------------------------------------------------
// 3) ev = exp(l - rowmax); denom = segment_sum(ev)   (ev overwrites logits)
// ---------------------------------------------------------------------------
__global__ void ev_kernel(const int* __restrict__ erow, float* __restrict__ logits,
                          const unsigned* __restrict__ rmax, float* __restrict__ denom, int e) {
    const int i = blockIdx.x * blockDim.x + threadIdx.x;
    if (i >= e) return;
    const int r = erow[i];
    const float m = sortable2f(rmax[r]);
    const float v = expf(logits[i] - m);
    logits[i] = v;
    atomicAdd(&denom[r], v);
}

// ---------------------------------------------------------------------------
// 4) SpMM scatter: one wave32 per edge, lane handles cols {lane, lane+32}
//    acc[row] += (ev/denom[row]) * fts[col]   (fts is L2-resident: 25.6 MB)
// ---------------------------------------------------------------------------
__global__ void spmm_kernel(const int* __restrict__ erow, const int* __restrict__ ecol,
                            const float* __restrict__ ev, const float* __restrict__ denom,
                            const float* __restrict__ fts, float* __restrict__ acc, int e) {
    const int gid  = blockIdx.x * blockDim.x + threadIdx.x;
    const int edge = gid >> 5;
    const int lane = threadIdx.x & 31;
    if (edge >= e) return;
    const int r = erow[edge], c = ecol[edge];
    const float coef = ev[edge] / denom[r];
    const float* __restrict__ src = fts + (size_t)c * F_OUT;
    float* __restrict__ dst = acc + (size_t)r * F_OUT;
    atomicAdd(&dst[lane],      coef * src[lane]);
    atomicAdd(&dst[lane + 32], coef * src[lane + 32]);
}

// ---------------------------------------------------------------------------
// 5) out = relu(acc + bias)   (in place on d_out)
// ---------------------------------------------------------------------------
__global__ void finalize_kernel(float* __restrict__ acc, const float* __restrict__ bias, int n) {
    const int i = blockIdx.x * blockDim.x + threadIdx.x;
    if (i >= n * F_OUT) return;
    const float v = acc[i] + bias[i & (F_OUT - 1)];
    acc[i] = v > 0.0f ? v : 0.0f;
}

// ---------------------------------------------------------------------------
extern "C" void kernel_launch(void* const* d_in, const int* in_sizes, int n_in,
                              void* d_out, int out_size, void* d_ws, size_t ws_size,
                              hipStream_t stream) {
    const float* seq  = (const float*)d_in[0];
    const int*   erow = (const int*)  d_in[1];
    const int*   ecol = (const int*)  d_in[2];
    const float* W    = (const float*)d_in[3];
    const float* a1   = (const float*)d_in[4];
    const float* b1   = (const float*)d_in[5];
    const float* a2   = (const float*)d_in[6];
    const float* b2   = (const float*)d_in[7];
    const float* bias = (const float*)d_in[8];

    const int n = in_sizes[0] / F_IN;    // 100000
    const int e = in_sizes[1];           // 3200000
    float* out = (float*)d_out;

    // workspace layout (f32 units)
    float*    fts   = (float*)d_ws;                    // n * 64
    float*    f1    = fts + (size_t)n * F_OUT;         // n
    float*    f2    = f1 + n;                          // n
    unsigned* rmax  = (unsigned*)(f2 + n);             // n
    float*    denom = (float*)(rmax + n);              // n
    float*    ev    = denom + n;                       // e

    // 0) init accumulator / segment state
    {
        const int tot = n * F_OUT;
        init_kernel<<<(tot + 255) / 256, 256, 0, stream>>>(out, rmax, denom, n);
    }
    // 1) WMMA GEMM + fused f1/f2: 8 waves/block, block covers 128 rows x 64 cols
    {
        const int blocks = (n + 127) / 128;            // 782
        gemm_wmma_kernel<<<blocks, 256, 0, stream>>>(seq, W, fts, a1, b1, a2, b2,
                                                     f1, f2, n);
    }
    // 2) edge logits + segment max
    logits_kernel<<<(e + 255) / 256, 256, 0, stream>>>(erow, ecol, f1, f2, ev, rmax, e);
    // 3) exp + segment sum
    ev_kernel<<<(e + 255) / 256, 256, 0, stream>>>(erow, ev, rmax, denom, e);
    // 4) SpMM scatter: 8 edges per 256-thread block
    {
        const long long threads = (long long)e * 32;
        const int blocks = (int)((threads + 255) / 256);
        spmm_kernel<<<blocks, 256, 0, stream>>>(erow, ecol, ev, denom, fts, out, e);
    }
    // 5) bias + relu
    finalize_kernel<<<(n * F_OUT + 255) / 256, 256, 0, stream>>>(out, bias, n);
}